// ClDice_20134806683698
// MI455X (gfx1250) — compile-verified
//
#include <hip/hip_runtime.h>
#include <cstdint>

#define BATCH 8
#define H 1024
#define W 1024
#define TILE 64
#define NPIX (BATCH * H * W)

typedef float v2f __attribute__((ext_vector_type(2)));
typedef float v8f __attribute__((ext_vector_type(8)));
typedef unsigned int u32x4 __attribute__((ext_vector_type(4)));
typedef unsigned int u32x8 __attribute__((ext_vector_type(8)));

// ---------- CDNA5 async global->LDS helpers ----------
__device__ __forceinline__ unsigned lds_off(const void* p) {
    // addrspacecast(LDS)->flat puts the LDS byte address in the low 32 bits
    return (unsigned)(unsigned long long)(uintptr_t)p;
}
__device__ __forceinline__ void async_load_f32(unsigned lds, unsigned long long gaddr) {
    asm volatile("global_load_async_to_lds_b32 %0, %1, off"
                 :: "v"(lds), "v"(gaddr) : "memory");
}
__device__ __forceinline__ void wait_async() {
    asm volatile("s_wait_asynccnt 0x0" ::: "memory");
}

// ---------- CDNA5 Tensor Data Mover: one 2D f32 tile -> LDS ----------
// Dense row-major landing in LDS (no padding): tileW x tileH elements,
// global row stride = strideElems. Caller must guarantee the whole tile is
// inside the image (TDM OOB returns zero, wrong identity for min/max).
__device__ __forceinline__ void tdm_load_tile_f32(const float* gsrc, unsigned lds_byte,
                                                  unsigned tileW, unsigned tileH,
                                                  unsigned strideElems)
{
    unsigned long long ga = (unsigned long long)(uintptr_t)gsrc;
    unsigned ga_lo = __builtin_amdgcn_readfirstlane((unsigned)(ga & 0xFFFFFFFFull));
    unsigned ga_hi = __builtin_amdgcn_readfirstlane((unsigned)(ga >> 32));
    unsigned lds_u = __builtin_amdgcn_readfirstlane(lds_byte);

    u32x4 g0;
    g0[0] = 1u;                                      // count=1, is_restore=0, gather off
    g0[1] = lds_u;                                   // lds_addr (bytes)
    g0[2] = ga_lo;                                   // global_addr[31:0]
    g0[3] = (ga_hi & 0x01FFFFFFu) | (2u << 30);      // global_addr[56:32] | type=2

    u32x8 g1;
    g1[0] = (2u << 16);                              // workgroup_mask=0, data_size=4B
    g1[1] = (tileW & 0xFFFFu) << 16;                 // tensor_dim0[15:0]
    g1[2] = (tileH & 0xFFFFu) << 16;                 // tensor_dim0[31:16]=0 | tensor_dim1[15:0]
    g1[3] = (tileW & 0xFFFFu) << 16;                 // tensor_dim1[31:16]=0 | tile_dim0
    g1[4] = (tileH & 0xFFFFu);                       // tile_dim1 | tile_dim2=0 (2D)
    g1[5] = strideElems;                             // tensor_dim0_stride[31:0]
    g1[6] = 0u;                                      // stride hi / dim1_stride lo (unused 2D)
    g1[7] = 0u;
    asm volatile("tensor_load_to_lds %0, %1" :: "s"(g0), "s"(g1) : "memory");
}

// ---------- WMMA-based exact f32 sum: total of all 64 A-elements per call ----------
__device__ __forceinline__ float wmma_col0_total(const v8f& c) {
    float s = 0.f;
#pragma unroll
    for (int k = 0; k < 8; ++k) s += c[k];
    // column j of D lives in lanes j and j+16 (8 VGPRs each); every column equals the total
    return __shfl(s, 0, 32) + __shfl(s, 16, 32);
}

// ============ Phase 1: sigmoid + materialize P/T + dice sums (WMMA reduction) ============
__global__ __launch_bounds__(256) void dice_reduce_kernel(
    const float* __restrict__ logits, const int* __restrict__ tgt,
    float* __restrict__ P, float* __restrict__ Tf, float* __restrict__ partials)
{
    const int tid = threadIdx.x;
    const int g   = blockIdx.x * 256 + tid;
    const int tot = gridDim.x * 256;
    v8f cI = {0,0,0,0,0,0,0,0};
    v8f cC = {0,0,0,0,0,0,0,0};
    v8f cT = {0,0,0,0,0,0,0,0};
    const v2f ones = {1.f, 1.f};
    for (int base = 0; base < NPIX; base += 2 * tot) {
        int e0 = base + g, e1 = base + tot + g;
        float p0 = 1.f / (1.f + expf(-logits[e0]));
        float p1 = 1.f / (1.f + expf(-logits[e1]));
        float t0 = (float)tgt[e0];
        float t1 = (float)tgt[e1];
        P[e0] = p0; P[e1] = p1; Tf[e0] = t0; Tf[e1] = t1;
        v2f aI = {p0 * t0, p1 * t1};
        v2f aC = {p0 + t0, p1 + t1};
        v2f aT = {t0, t1};
        cI = __builtin_amdgcn_wmma_f32_16x16x4_f32(false, aI, false, ones, (short)0, cI, false, false);
        cC = __builtin_amdgcn_wmma_f32_16x16x4_f32(false, aC, false, ones, (short)0, cC, false, false);
        cT = __builtin_amdgcn_wmma_f32_16x16x4_f32(false, aT, false, ones, (short)0, cT, false, false);
    }
    float sI = wmma_col0_total(cI);
    float sC = wmma_col0_total(cC);
    float sT = wmma_col0_total(cT);
    __shared__ float wsum[8][3];
    int wave = tid >> 5, lane = tid & 31;
    if (lane == 0) { wsum[wave][0] = sI; wsum[wave][1] = sC; wsum[wave][2] = sT; }
    __syncthreads();
    if (tid == 0) {
        float a = 0, b = 0, c = 0;
        for (int w2 = 0; w2 < 8; ++w2) { a += wsum[w2][0]; b += wsum[w2][1]; c += wsum[w2][2]; }
        partials[blockIdx.x * 3 + 0] = a;
        partials[blockIdx.x * 3 + 1] = b;
        partials[blockIdx.x * 3 + 2] = c;
    }
}

// ============ clDice sums over the two skeletons (WMMA reduction) ============
__global__ __launch_bounds__(256) void cl_reduce_kernel(
    const float* __restrict__ A, const float* __restrict__ B, float* __restrict__ partials)
{
    const int tid = threadIdx.x;
    const int g   = blockIdx.x * 256 + tid;
    const int tot = gridDim.x * 256;
    v8f cI = {0,0,0,0,0,0,0,0};
    v8f cA = {0,0,0,0,0,0,0,0};
    v8f cB = {0,0,0,0,0,0,0,0};
    const v2f ones = {1.f, 1.f};
    for (int base = 0; base < NPIX; base += 2 * tot) {
        int e0 = base + g, e1 = base + tot + g;
        float a0 = A[e0], a1 = A[e1];
        float b0 = B[e0], b1 = B[e1];
        v2f vI = {a0 * b0, a1 * b1};
        v2f vA = {a0, a1};
        v2f vB = {b0, b1};
        cI = __builtin_amdgcn_wmma_f32_16x16x4_f32(false, vI, false, ones, (short)0, cI, false, false);
        cA = __builtin_amdgcn_wmma_f32_16x16x4_f32(false, vA, false, ones, (short)0, cA, false, false);
        cB = __builtin_amdgcn_wmma_f32_16x16x4_f32(false, vB, false, ones, (short)0, cB, false, false);
    }
    float sI = wmma_col0_total(cI);
    float sA = wmma_col0_total(cA);
    float sB = wmma_col0_total(cB);
    __shared__ float wsum[8][3];
    int wave = tid >> 5, lane = tid & 31;
    if (lane == 0) { wsum[wave][0] = sI; wsum[wave][1] = sA; wsum[wave][2] = sB; }
    __syncthreads();
    if (tid == 0) {
        float a = 0, b = 0, c = 0;
        for (int w2 = 0; w2 < 8; ++w2) { a += wsum[w2][0]; b += wsum[w2][1]; c += wsum[w2][2]; }
        partials[blockIdx.x * 3 + 0] = a;
        partials[blockIdx.x * 3 + 1] = b;
        partials[blockIdx.x * 3 + 2] = c;
    }
}

// ============ deterministic partial reduction (single block) ============
__global__ __launch_bounds__(256) void reduce_partials_kernel(
    const float* __restrict__ partials, int n, float* __restrict__ out3)
{
    __shared__ float sa[256], sb[256], sc[256];
    int tid = threadIdx.x;
    float a = 0, b = 0, c = 0;
    for (int i = tid; i < n; i += 256) {
        a += partials[3 * i + 0];
        b += partials[3 * i + 1];
        c += partials[3 * i + 2];
    }
    sa[tid] = a; sb[tid] = b; sc[tid] = c;
    __syncthreads();
    for (int s = 128; s > 0; s >>= 1) {
        if (tid < s) { sa[tid] += sa[tid + s]; sb[tid] += sb[tid + s]; sc[tid] += sc[tid + s]; }
        __syncthreads();
    }
    if (tid == 0) { out3[0] = sa[0]; out3[1] = sb[0]; out3[2] = sc[0]; }
}

// ============ skeleton init: skel = relu(im - dilate(erode(im))) ============
// erode = 5-point cross min (OOB = +inf); dilate = 3x3 max (OOB = -inf)
__global__ __launch_bounds__(256) void skel_init_kernel(
    const float* __restrict__ im, float* __restrict__ skel)
{
    __shared__ float sIm[68 * 68];  // halo 2
    __shared__ float sE [66 * 66];  // halo 1
    const int b   = blockIdx.z;
    const int ty0 = blockIdx.y * TILE, tx0 = blockIdx.x * TILE;
    const int tid = threadIdx.y * 16 + threadIdx.x;
    const float* imb = im + (size_t)b * (H * W);

    // Whole 68x68 halo window inside the image? -> single TDM descriptor.
    const bool interior = (ty0 >= 2) && (ty0 + 66 <= H) && (tx0 >= 2) && (tx0 + 66 <= W);
    if (interior) {
        if (tid < 32) {
            if (tid == 0)
                tdm_load_tile_f32(imb + (size_t)(ty0 - 2) * W + (tx0 - 2),
                                  lds_off(&sIm[0]), 68u, 68u, (unsigned)W);
            __builtin_amdgcn_s_wait_tensorcnt(0);
        }
    } else {
        for (int i = tid; i < 68 * 68; i += 256) {
            int r = i / 68, c = i - r * 68;
            int y = ty0 - 2 + r, x = tx0 - 2 + c;
            if ((unsigned)y < H && (unsigned)x < W)
                async_load_f32(lds_off(&sIm[i]),
                               (unsigned long long)(uintptr_t)(imb + (size_t)y * W + x));
            else
                sIm[i] = __builtin_inff();   // min-identity halo
        }
        wait_async();
    }
    __syncthreads();

    for (int i = tid; i < 66 * 66; i += 256) {
        int r = i / 66, c = i - r * 66;
        int y = ty0 - 1 + r, x = tx0 - 1 + c;
        float e;
        if ((unsigned)y < H && (unsigned)x < W) {
            int ci = (r + 1) * 68 + (c + 1);
            float m = sIm[ci];
            m = fminf(m, sIm[ci - 68]); m = fminf(m, sIm[ci + 68]);
            m = fminf(m, sIm[ci - 1]);  m = fminf(m, sIm[ci + 1]);
            e = m;
        } else {
            e = -__builtin_inff();       // max-identity for the dilate
        }
        sE[i] = e;
    }
    __syncthreads();

#pragma unroll
    for (int j = 0; j < 4; ++j)
#pragma unroll
        for (int i2 = 0; i2 < 4; ++i2) {
            int yy = threadIdx.y + 16 * j, xx = threadIdx.x + 16 * i2;
            int ci = (yy + 1) * 66 + (xx + 1);
            float M = sE[ci];
            M = fmaxf(M, sE[ci - 1]);  M = fmaxf(M, sE[ci + 1]);
            M = fmaxf(M, sE[ci - 67]); M = fmaxf(M, sE[ci - 66]); M = fmaxf(M, sE[ci - 65]);
            M = fmaxf(M, sE[ci + 65]); M = fmaxf(M, sE[ci + 66]); M = fmaxf(M, sE[ci + 67]);
            float v = sIm[(yy + 2) * 68 + (xx + 2)];
            size_t gidx = (size_t)b * (H * W) + (size_t)(ty0 + yy) * W + (tx0 + xx);
            skel[gidx] = fmaxf(v - M, 0.f);
        }
}

// ============ one skeleton iteration, fully fused ============
// E  = erode(im)           -> new im
// E2 = erode(E)
// O  = dilate3x3(E2)
// delta = relu(E - O); skel += relu(delta - skel*delta)
__global__ __launch_bounds__(256) void skel_iter_kernel(
    const float* __restrict__ im, float* __restrict__ imOut, float* __restrict__ skel)
{
    __shared__ float sIm[70 * 70];  // halo 3
    __shared__ float sE [68 * 68];  // halo 2
    __shared__ float sE2[66 * 66];  // halo 1
    const int b   = blockIdx.z;
    const int ty0 = blockIdx.y * TILE, tx0 = blockIdx.x * TILE;
    const int tid = threadIdx.y * 16 + threadIdx.x;
    const float* imb = im + (size_t)b * (H * W);

    // Whole 70x70 halo window inside the image? -> single TDM descriptor.
    const bool interior = (ty0 >= 3) && (ty0 + 67 <= H) && (tx0 >= 3) && (tx0 + 67 <= W);
    if (interior) {
        if (tid < 32) {
            if (tid == 0)
                tdm_load_tile_f32(imb + (size_t)(ty0 - 3) * W + (tx0 - 3),
                                  lds_off(&sIm[0]), 70u, 70u, (unsigned)W);
            __builtin_amdgcn_s_wait_tensorcnt(0);
        }
    } else {
        for (int i = tid; i < 70 * 70; i += 256) {
            int r = i / 70, c = i - r * 70;
            int y = ty0 - 3 + r, x = tx0 - 3 + c;
            if ((unsigned)y < H && (unsigned)x < W)
                async_load_f32(lds_off(&sIm[i]),
                               (unsigned long long)(uintptr_t)(imb + (size_t)y * W + x));
            else
                sIm[i] = __builtin_inff();
        }
        wait_async();
    }
    __syncthreads();

    for (int i = tid; i < 68 * 68; i += 256) {
        int r = i / 68, c = i - r * 68;
        int y = ty0 - 2 + r, x = tx0 - 2 + c;
        float e;
        if ((unsigned)y < H && (unsigned)x < W) {
            int ci = (r + 1) * 70 + (c + 1);
            float m = sIm[ci];
            m = fminf(m, sIm[ci - 70]); m = fminf(m, sIm[ci + 70]);
            m = fminf(m, sIm[ci - 1]);  m = fminf(m, sIm[ci + 1]);
            e = m;
        } else {
            e = __builtin_inff();        // min-identity for next erode
        }
        sE[i] = e;
    }
    __syncthreads();

    for (int i = tid; i < 66 * 66; i += 256) {
        int r = i / 66, c = i - r * 66;
        int y = ty0 - 1 + r, x = tx0 - 1 + c;
        float e;
        if ((unsigned)y < H && (unsigned)x < W) {
            int ci = (r + 1) * 68 + (c + 1);
            float m = sE[ci];
            m = fminf(m, sE[ci - 68]); m = fminf(m, sE[ci + 68]);
            m = fminf(m, sE[ci - 1]);  m = fminf(m, sE[ci + 1]);
            e = m;
        } else {
            e = -__builtin_inff();       // max-identity for the dilate
        }
        sE2[i] = e;
    }
    __syncthreads();

#pragma unroll
    for (int j = 0; j < 4; ++j)
#pragma unroll
        for (int i2 = 0; i2 < 4; ++i2) {
            int yy = threadIdx.y + 16 * j, xx = threadIdx.x + 16 * i2;
            int ci = (yy + 1) * 66 + (xx + 1);
            float M = sE2[ci];
            M = fmaxf(M, sE2[ci - 1]);  M = fmaxf(M, sE2[ci + 1]);
            M = fmaxf(M, sE2[ci - 67]); M = fmaxf(M, sE2[ci - 66]); M = fmaxf(M, sE2[ci - 65]);
            M = fmaxf(M, sE2[ci + 65]); M = fmaxf(M, sE2[ci + 66]); M = fmaxf(M, sE2[ci + 67]);
            float e = sE[(yy + 2) * 68 + (xx + 2)];   // eroded image = new im
            float delta = fmaxf(e - M, 0.f);
            size_t gidx = (size_t)b * (H * W) + (size_t)(ty0 + yy) * W + (tx0 + xx);
            float sk = skel[gidx];
            sk = sk + fmaxf(delta - sk * delta, 0.f);
            skel[gidx] = sk;
            imOut[gidx] = e;
        }
}

// ============ final scalar combine ============
__global__ void final_kernel(const float* __restrict__ sums, float* __restrict__ out)
{
    float I = sums[0], C = sums[1], T = sums[2];
    float score = (2.f * I + 1.f) / fmaxf(C + 1.f, 1e-07f);
    float dice  = (1.f - score) * (T > 0.f ? 1.f : 0.f);
    float Ic = sums[3], So = sums[4], St = sums[5];
    float tprec = (Ic + 1.f) / (So + 1.f);
    float tsens = (Ic + 1.f) / (St + 1.f);
    float cls = 2.f * (tprec * tsens) / (tprec + tsens);
    float cl  = (1.f - cls) * (St > 0.f ? 1.f : 0.f);
    out[0] = 0.5f * dice + 0.5f * cl;
}

extern "C" void kernel_launch(void* const* d_in, const int* in_sizes, int n_in,
                              void* d_out, int out_size, void* d_ws, size_t ws_size,
                              hipStream_t stream) {
    (void)in_sizes; (void)n_in; (void)out_size; (void)ws_size;
    const float* logits = (const float*)d_in[0];
    const int*   target = (const int*)d_in[1];
    float* out = (float*)d_out;
    float* w   = (float*)d_ws;

    const size_t N = (size_t)NPIX;
    float* P    = w;                 // probs -> im ping (field o)
    float* Bo   = w + 1 * N;         // im pong (field o)
    float* Ko   = w + 2 * N;         // skel_o
    float* Tf   = w + 3 * N;         // target float -> im ping (field t)
    float* Bt   = w + 4 * N;         // im pong (field t)
    float* Kt   = w + 5 * N;         // skel_t
    float* part = w + 6 * N;         // 1024*3 partials (reused)
    float* sums = w + 6 * N + 4096;  // [0..2] dice sums, [3..5] cl sums

    dice_reduce_kernel<<<dim3(1024), dim3(256), 0, stream>>>(logits, target, P, Tf, part);
    reduce_partials_kernel<<<dim3(1), dim3(256), 0, stream>>>(part, 1024, sums);

    dim3 sg(W / TILE, H / TILE, BATCH), sb(16, 16, 1);

    skel_init_kernel<<<sg, sb, 0, stream>>>(P, Ko);
    {
        float* ping = P; float* pong = Bo;
        for (int it = 0; it < 10; ++it) {
            skel_iter_kernel<<<sg, sb, 0, stream>>>(ping, pong, Ko);
            float* t = ping; ping = pong; pong = t;
        }
    }

    skel_init_kernel<<<sg, sb, 0, stream>>>(Tf, Kt);
    {
        float* ping = Tf; float* pong = Bt;
        for (int it = 0; it < 10; ++it) {
            skel_iter_kernel<<<sg, sb, 0, stream>>>(ping, pong, Kt);
            float* t = ping; ping = pong; pong = t;
        }
    }

    cl_reduce_kernel<<<dim3(1024), dim3(256), 0, stream>>>(Ko, Kt, part);
    reduce_partials_kernel<<<dim3(1), dim3(256), 0, stream>>>(part, 1024, sums + 3);
    final_kernel<<<dim3(1), dim3(1), 0, stream>>>(sums, out);
}